// LatticeLagrangian_41446434406802
// MI455X (gfx1250) — compile-verified
//
#include <hip/hip_runtime.h>
#include <math.h>

// ---------------------------------------------------------------------------
// LatticeLagrangian on gfx1250 (MI455X, CDNA5, wave32)
//
// V2 = sum_{i != j} 1/r_ij via 16x16 pair tiles with V_WMMA_F32_16X16X4_F32.
//
//   A row m = (-s2*x_m, -s2*y_m, -s2*z_m, 1)       (s2 = sqrt(2), setup once)
//   B col n = ( s2*x_n,  s2*y_n,  s2*z_n, |x_n|^2)  (pre-swizzled per lane)
//   C[m][n] = |x_m|^2                               (j-invariant, hoisted)
//   => D = |x_m|^2 + |x_n|^2 - 2*x_m.x_n = r2       (no post-scaling at all)
//
// grid (512, 2) x 8 waves: wave sweeps residue class rc = wave + 8*by mod 16
// with two linear loops (above/below diagonal), software-pipelined B loads
// (bsw padded by 16 tiles so prefetch needs no guards). Clamp via single
// v_med3_f32 (finite upper bound prevents the maxnum fold). Diagonal tile
// peeled; executed by ALL waves with full EXEC (WMMA requirement),
// accumulated only by rc==15 lanes. One atomic per block via LDS reduce.
// ---------------------------------------------------------------------------

#define NATOMS 8192
#define NFREE  4096
#define NTILES (NATOMS / 16)   // 512
#define RSTEP  16              // residue classes mod 16
#define BPAD   16              // padding tiles for unguarded prefetch

typedef __attribute__((ext_vector_type(2))) float v2f;
typedef __attribute__((ext_vector_type(8))) float v8f;

#define SQRT2F  1.41421356237309504880f
#define FLT_BIG 3.402823466e38f

// ---- out[0] = 0 -----------------------------------------------------------
__global__ void ll_init_out(float* out) { out[0] = 0.0f; }

// ---- xfull[i] = (x, y, z, |x|^2) from x0 -----------------------------------
__global__ void ll_build_x(const float* __restrict__ x0,
                           float4* __restrict__ xfull, int n) {
  int i = blockIdx.x * blockDim.x + threadIdx.x;
  if (i < n) {
    float x = x0[3 * i + 0];
    float y = x0[3 * i + 1];
    float z = x0[3 * i + 2];
    float4 v;
    v.x = x;
    v.y = y;
    v.z = z;
    v.w = x * x + y * y + z * z;
    xfull[i] = v;
  }
}

// ---- scatter free coordinates q into xfull via free_idx --------------------
__global__ void ll_scatter_q(const float* __restrict__ q,
                             const int* __restrict__ free_idx,
                             float4* __restrict__ xfull, int nfree) {
  int k = blockIdx.x * blockDim.x + threadIdx.x;
  if (k < nfree) {
    int i = free_idx[k];
    float x = q[3 * k + 0];
    float y = q[3 * k + 1];
    float z = q[3 * k + 2];
    float4 v;
    v.x = x;
    v.y = y;
    v.z = z;
    v.w = x * x + y * y + z * z;
    xfull[i] = v;
  }
}

// ---- pre-swizzled B operand: bsw[tile*32 + lane] ---------------------------
// lanes 0-15  -> {K0,K1} = ( s2*x, s2*y ) of atom tile*16+lane
// lanes 16-31 -> {K2,K3} = ( s2*z, |x|^2) of atom tile*16+lane-16
__global__ void ll_build_bsw(const float4* __restrict__ xfull,
                             v2f* __restrict__ bsw, int nlanes) {
  int t = blockIdx.x * blockDim.x + threadIdx.x;
  if (t < nlanes) {
    int tile = t >> 5;
    int l    = t & 31;
    const float4 v = xfull[tile * 16 + (l & 15)];
    v2f b;
    if (l < 16) {
      b.x = SQRT2F * v.x;
      b.y = SQRT2F * v.y;
    } else {
      b.x = SQRT2F * v.z;
      b.y = v.w;
    }
    bsw[t] = b;
  }
}

// ---- T = 0.5 * sum(qd^2) accumulated into out ------------------------------
__global__ void ll_kinetic(const float* __restrict__ qd, float* out, int n) {
  float s = 0.0f;
  for (int i = blockIdx.x * blockDim.x + threadIdx.x; i < n;
       i += gridDim.x * blockDim.x) {
    float v = qd[i];
    s += v * v;
  }
#pragma unroll
  for (int off = 16; off > 0; off >>= 1) s += __shfl_xor(s, off, 32);
  if ((threadIdx.x & 31) == 0) atomicAdd(out, 0.5f * s);
}

// ---- pair kernel: out -= sum_{i != j} 1/r_ij -------------------------------
__global__ __launch_bounds__(256) void ll_pairs(
    const float4* __restrict__ xfull, const v2f* __restrict__ bsw,
    float* __restrict__ out) {
  const int lane   = threadIdx.x & 31;
  const int wave   = threadIdx.x >> 5;
  const int lane15 = lane & 15;
  const bool hi    = lane >= 16;
  const int diag   = blockIdx.x;            // this block's i-tile
  const int rc     = wave + 8 * blockIdx.y; // residue class 0..15
  const int i0     = diag * 16;

  // A tile (16x4 f32, 2 VGPRs): K = (-s2*x, -s2*y, -s2*z, 1).
  const float4 va = xfull[i0 + lane15];
  v2f a;
  a.x = hi ? (-SQRT2F * va.z) : (-SQRT2F * va.x);
  a.y = hi ? 1.0f             : (-SQRT2F * va.y);

  // C = |x_i|^2, constant over the whole j sweep (M = v + hi*8).
  v8f cc;
#pragma unroll
  for (int v = 0; v < 8; ++v) cc[v] = xfull[i0 + v + (hi ? 8 : 0)].w;

  float acc0 = 0.0f, acc1 = 0.0f;

  const int s1 = diag + 1 + rc;  // class start above the diagonal (< 528)

  // ---- tiles ABOVE the diagonal: jt = s1, s1+16, ... < NTILES -------------
  {
    v2f b = bsw[s1 * 32 + lane];                    // prologue (padded)
    for (int jt = s1; jt < NTILES; jt += RSTEP) {
      const v2f bn = bsw[(jt + RSTEP) * 32 + lane]; // prefetch (padded)
      v8f d = __builtin_amdgcn_wmma_f32_16x16x4_f32(
          false, a, false, b, (short)0, cc, false, false);
#pragma unroll
      for (int v = 0; v < 8; v += 2) {
        acc0 += __builtin_amdgcn_rsqf(
            __builtin_amdgcn_fmed3f(d[v],     1e-12f, FLT_BIG));
        acc1 += __builtin_amdgcn_rsqf(
            __builtin_amdgcn_fmed3f(d[v + 1], 1e-12f, FLT_BIG));
      }
      b = bn;
    }
  }

  // ---- tiles BELOW the diagonal (wrapped): jt = s1%16, ... < diag ---------
  {
    const int r0 = s1 & (RSTEP - 1);
    v2f b = bsw[r0 * 32 + lane];                    // prologue
    for (int jt = r0; jt < diag; jt += RSTEP) {
      const v2f bn = bsw[(jt + RSTEP) * 32 + lane]; // prefetch (in range)
      v8f d = __builtin_amdgcn_wmma_f32_16x16x4_f32(
          false, a, false, b, (short)0, cc, false, false);
#pragma unroll
      for (int v = 0; v < 8; v += 2) {
        acc0 += __builtin_amdgcn_rsqf(
            __builtin_amdgcn_fmed3f(d[v],     1e-12f, FLT_BIG));
        acc1 += __builtin_amdgcn_rsqf(
            __builtin_amdgcn_fmed3f(d[v + 1], 1e-12f, FLT_BIG));
      }
      b = bn;
    }
  }

  // ---- diagonal tile: all waves execute (full-EXEC WMMA); only residue
  //      class 15 accumulates, excluding true diagonal entries. -------------
  {
    const v2f b = bsw[diag * 32 + lane];
    v8f d = __builtin_amdgcn_wmma_f32_16x16x4_f32(
        false, a, false, b, (short)0, cc, false, false);
#pragma unroll
    for (int v = 0; v < 8; ++v) {
      float r = __builtin_amdgcn_rsqf(
          __builtin_amdgcn_fmed3f(d[v], 1e-12f, FLT_BIG));
      bool keep = (rc == 15) && ((v + (hi ? 8 : 0)) != lane15);
      acc0 += keep ? r : 0.0f;
    }
  }

  // ---- reduction: wave32 shuffle, then LDS across 8 waves, 1 atomic/block -
  float acc = acc0 + acc1;
#pragma unroll
  for (int off = 16; off > 0; off >>= 1) acc += __shfl_xor(acc, off, 32);

  __shared__ float wsum[8];
  if (lane == 0) wsum[wave] = acc;
  __syncthreads();
  if (threadIdx.x == 0) {
    float t = 0.0f;
#pragma unroll
    for (int w = 0; w < 8; ++w) t += wsum[w];
    atomicAdd(out, -t);
  }
}

// ---------------------------------------------------------------------------
extern "C" void kernel_launch(void* const* d_in, const int* in_sizes, int n_in,
                              void* d_out, int out_size, void* d_ws,
                              size_t ws_size, hipStream_t stream) {
  const float* q        = (const float*)d_in[0];  // [NFREE,3]
  const float* qd       = (const float*)d_in[1];  // [NFREE,3]
  const float* x0       = (const float*)d_in[2];  // [NATOMS,3]
  const int*   free_idx = (const int*)d_in[3];    // [NFREE]
  float* out = (float*)d_out;

  float4* xfull = (float4*)d_ws;                  // 128 KB
  v2f*    bsw   = (v2f*)(xfull + NATOMS);         // (512+16) tiles, ~135 KB

  ll_init_out<<<1, 1, 0, stream>>>(out);
  ll_build_x<<<(NATOMS + 255) / 256, 256, 0, stream>>>(x0, xfull, NATOMS);
  ll_scatter_q<<<(NFREE + 255) / 256, 256, 0, stream>>>(q, free_idx, xfull,
                                                        NFREE);
  ll_build_bsw<<<(NTILES * 32 + 255) / 256, 256, 0, stream>>>(
      xfull, bsw, NTILES * 32);
  ll_kinetic<<<12, 256, 0, stream>>>(qd, out, NFREE * 3);
  ll_pairs<<<dim3(NTILES, 2), 256, 0, stream>>>(xfull, bsw, out);
}